// Model_67293547594175
// MI455X (gfx1250) — compile-verified
//
#include <hip/hip_runtime.h>
#include <hip/hip_bf16.h>
#include <stdint.h>

// MRoPE (multimodal RoPE), HEAD_SIZE = ROTARY_DIM = 128, sections (32,32,32).
// Pure bandwidth kernel: ~690 MB moved -> ~30 us at 23.3 TB/s. No matrix op,
// so optimization = b128 vector memory ops, NT hints on the streamed q/k data,
// and gfx1250 async global->LDS staging of the per-token cos/sin rows.

typedef float v4f __attribute__((ext_vector_type(4)));

__global__ __launch_bounds__(256) void mrope_kernel(
    const float* __restrict__ qp, const float* __restrict__ kp,
    const float* __restrict__ cp, const float* __restrict__ sp,
    float* __restrict__ oq, float* __restrict__ ok,
    int T, int nqh, int nkh)
{
    // [0,64): cos row for this token, [64,128): sin row
    __shared__ __align__(16) float lds_cs[128];

    const int t   = blockIdx.x;
    const int tid = threadIdx.x;

    // ---- Stage cos/sin for token t into LDS with one wave32 async copy ----
    // lanes 0..15 : cos chunks 0..15 (chunks 0-7 from section 0, 8-15 from section 1)
    // lanes 16..31: sin chunks 0..15
    // 32 lanes x 16B = 512B total, tracked by ASYNCcnt.
    if (tid < 32) {
        const int isSin = tid >> 4;
        const int chunk = tid & 15;          // float4 chunk within the 64-wide row
        const int sec   = chunk >> 3;        // elements 0..31 -> cos[0], 32..63 -> cos[1]
        const float* src = (isSin ? sp : cp) + ((size_t)sec * T + t) * 64 + chunk * 4;
        float*       dst = lds_cs + isSin * 64 + chunk * 4;

        uint32_t ldsOff = (uint32_t)(uintptr_t)dst;  // low 32 bits of flat addr = LDS offset
        uint64_t gaddr  = (uint64_t)(uintptr_t)src;
        asm volatile("global_load_async_to_lds_b128 %0, %1, off"
                     :: "v"(ldsOff), "v"(gaddr) : "memory");
        asm volatile("s_wait_asynccnt 0x0" ::: "memory");
    }
    __syncthreads();

    // ---- Rotate all heads of this token ----
    // One task = one float4 pair: x1 at [h*128 + c*4], x2 at [h*128 + 64 + c*4].
    const int qTasks = nqh * 16;
    const int total  = qTasks + nkh * 16;

    const size_t qRow = (size_t)t * nqh * 128;
    const size_t kRow = (size_t)t * nkh * 128;

    for (int task = tid; task < total; task += 256) {
        const float* base;
        float*       obase;
        int          local;
        if (task < qTasks) {
            base  = qp + qRow;
            obase = oq + qRow;
            local = task;
        } else {
            base  = kp + kRow;
            obase = ok + kRow;
            local = task - qTasks;
        }
        const int h = local >> 4;
        const int c = local & 15;

        const float* p1 = base + h * 128 + c * 4;        // x1 chunk
        const float* p2 = p1 + 64;                       // x2 chunk

        v4f x1 = __builtin_nontemporal_load((const v4f*)p1);
        v4f x2 = __builtin_nontemporal_load((const v4f*)p2);
        v4f cv = *(const v4f*)(lds_cs + c * 4);
        v4f sv = *(const v4f*)(lds_cs + 64 + c * 4);

        v4f n1 = x1 * cv - x2 * sv;
        v4f n2 = x2 * cv + x1 * sv;

        __builtin_nontemporal_store(n1, (v4f*)(obase + h * 128 + c * 4));
        __builtin_nontemporal_store(n2, (v4f*)(obase + h * 128 + 64 + c * 4));
    }
}

extern "C" void kernel_launch(void* const* d_in, const int* in_sizes, int n_in,
                              void* d_out, int out_size, void* d_ws, size_t ws_size,
                              hipStream_t stream) {
    const float* q  = (const float*)d_in[0];
    const float* k  = (const float*)d_in[1];
    const float* cp = (const float*)d_in[2];
    const float* sp = (const float*)d_in[3];

    // cos: (3, T, 64) -> T; q: (T, nqh*128); k: (T, nkh*128)
    const int T   = in_sizes[2] / (3 * 64);
    const int nqh = in_sizes[0] / (T * 128);
    const int nkh = in_sizes[1] / (T * 128);

    float* oq = (float*)d_out;
    float* ok = oq + (size_t)T * nqh * 128;

    dim3 grid(T);
    dim3 block(256);
    hipLaunchKernelGGL(mrope_kernel, grid, block, 0, stream,
                       q, k, cp, sp, oq, ok, T, nqh, nkh);
}